// DataReUploadingLinear_36094905155664
// MI455X (gfx1250) — compile-verified
//
#include <hip/hip_runtime.h>

typedef float v2f __attribute__((ext_vector_type(2)));
typedef float v8f __attribute__((ext_vector_type(8)));

#define NQ      6
#define DIM     64
#define NPAULI  4096
#define PITCH   66
#define MSZ     (DIM * PITCH)          // floats per real 64x64 buffer (padded)
#define LDS_FLOATS (6 * MSZ + 512)
#define HORNER_K 12

// LDS layout (float offsets)
#define OFF_MRE 0
#define OFF_MIM (1 * MSZ)
#define OFF_RRE (2 * MSZ)
#define OFF_RIM (3 * MSZ)
#define OFF_TRE (4 * MSZ)
#define OFF_TIM (5 * MSZ)
#define OFF_SCR (6 * MSZ)

__device__ __forceinline__ v8f wmma_f32(v2f a, v2f b, v8f c) {
    // D = A(16x4) * B(4x16) + C, fp32 WMMA (CDNA5)
    return __builtin_amdgcn_wmma_f32_16x16x4_f32(false, a, false, b, (short)0, c,
                                                 false, false);
}

// C = A * B for 64x64 complex matrices in LDS (row-major, PITCH stride).
// 8 waves; each wave computes a 16x32 slab (two adjacent 16x16 tiles sharing
// the same A fragments -> 6 LDS fragment loads per 8 WMMAs).
__device__ void cmatmul(const float* __restrict__ Are, const float* __restrict__ Aim,
                        const float* __restrict__ Bre, const float* __restrict__ Bim,
                        float* __restrict__ Cre, float* __restrict__ Cim) {
    const int lane = threadIdx.x & 31;
    const int wave = threadIdx.x >> 5;
    const int h    = lane >> 4;     // half-wave select (K pair)
    const int l16  = lane & 15;
    const int ti   = wave >> 1;            // tile row 0..3
    const int tj   = (wave & 1) * 2;       // tile col pair {0,1} or {2,3}
    const int row  = ti * 16 + l16;
    const int col0 = tj * 16 + l16;
    const int col1 = col0 + 16;

    v8f rr0 = {0,0,0,0,0,0,0,0}, ii0 = {0,0,0,0,0,0,0,0};
    v8f ri0 = {0,0,0,0,0,0,0,0}, ir0 = {0,0,0,0,0,0,0,0};
    v8f rr1 = {0,0,0,0,0,0,0,0}, ii1 = {0,0,0,0,0,0,0,0};
    v8f ri1 = {0,0,0,0,0,0,0,0}, ir1 = {0,0,0,0,0,0,0,0};

    for (int kk = 0; kk < DIM; kk += 4) {
        const int ka = kk + 2 * h;
        v2f ar  = { Are[row * PITCH + ka], Are[row * PITCH + ka + 1] };
        v2f ai  = { Aim[row * PITCH + ka], Aim[row * PITCH + ka + 1] };
        v2f br0 = { Bre[ka * PITCH + col0], Bre[(ka + 1) * PITCH + col0] };
        v2f bi0 = { Bim[ka * PITCH + col0], Bim[(ka + 1) * PITCH + col0] };
        v2f br1 = { Bre[ka * PITCH + col1], Bre[(ka + 1) * PITCH + col1] };
        v2f bi1 = { Bim[ka * PITCH + col1], Bim[(ka + 1) * PITCH + col1] };
        rr0 = wmma_f32(ar, br0, rr0);
        ii0 = wmma_f32(ai, bi0, ii0);
        ri0 = wmma_f32(ar, bi0, ri0);
        ir0 = wmma_f32(ai, br0, ir0);
        rr1 = wmma_f32(ar, br1, rr1);
        ii1 = wmma_f32(ai, bi1, ii1);
        ri1 = wmma_f32(ar, bi1, ri1);
        ir1 = wmma_f32(ai, br1, ir1);
    }
    const int r0 = ti * 16 + h * 8;
    #pragma unroll
    for (int v = 0; v < 8; ++v) {
        Cre[(r0 + v) * PITCH + col0] = rr0[v] - ii0[v];   // Re = ArBr - AiBi
        Cim[(r0 + v) * PITCH + col0] = ri0[v] + ir0[v];   // Im = ArBi + AiBr
        Cre[(r0 + v) * PITCH + col1] = rr1[v] - ii1[v];
        Cim[(r0 + v) * PITCH + col1] = ri1[v] + ir1[v];
    }
}

// Build M = -i * H where H = sum_p c_p P_p, directly into LDS M buffers.
// permL[xm*64+zm] = base-4 Pauli index p for that (x-mask, z-mask) pair (LDS).
// coef index = p - shift ; out-of-range coefficients are zero.
__device__ void build_M(float* lds, const unsigned* __restrict__ permL,
                        const float* __restrict__ coef, int shift, int ncoef) {
    float* Mre = lds + OFF_MRE;
    float* Mim = lds + OFF_MIM;
    const int tid = threadIdx.x;
    const int i  = tid & 63;    // matrix row owned by this thread
    const int qq = tid >> 6;    // xmask quarter
    for (int xl = 0; xl < 16; ++xl) {
        const int xm = qq * 16 + xl;
        const int j = i ^ xm;
        float aRe = 0.f, aIm = 0.f;
        for (int zm = 0; zm < DIM; ++zm) {
            const unsigned p   = permL[xm * DIM + zm];
            const unsigned idx = p - (unsigned)shift;
            float c = (idx < (unsigned)ncoef) ? coef[idx] : 0.0f;
            const int nY = __popc(xm & zm) & 3;             // (-i)^nY phase
            float v = (__popc(zm & i) & 1) ? -c : c;        // (-1)^{popc(zm&i)}
            v = (nY >= 2) ? v : -v;
            if (nY & 1) aRe += v;   // imaginary H part -> Re(-iH)
            else        aIm += v;   // real H part      -> Im(-iH) (negated above)
        }
        Mre[i * PITCH + j] = aRe;
        Mim[i * PITCH + j] = aIm;
    }
    __syncthreads();
}

// In-place expm of complex M (in LDS) via scaling-and-squaring + Horner-Taylor.
// Result left in R buffers.
__device__ void expm_lds(float* lds) {
    float* Mre = lds + OFF_MRE;  float* Mim = lds + OFF_MIM;
    float* Rre = lds + OFF_RRE;  float* Rim = lds + OFF_RIM;
    float* Tre = lds + OFF_TRE;  float* Tim = lds + OFF_TIM;
    float* scr = lds + OFF_SCR;
    const int tid = threadIdx.x;

    __syncthreads();
    // inf-norm bound: max row sum of |Re|+|Im|
    if (tid < DIM) {
        float s = 0.f;
        for (int j = 0; j < DIM; ++j)
            s += fabsf(Mre[tid * PITCH + j]) + fabsf(Mim[tid * PITCH + j]);
        scr[tid] = s;
    }
    __syncthreads();
    if (tid == 0) {
        float m = 0.f;
        for (int k = 0; k < DIM; ++k) m = fmaxf(m, scr[k]);
        int ns = 0;
        while (m > 0.5f && ns < 40) { m *= 0.5f; ++ns; }
        ((int*)scr)[64] = ns;
    }
    __syncthreads();
    const int nsq = ((int*)scr)[64];
    const float sc = ldexpf(1.0f, -nsq);

    for (int idx = tid; idx < DIM * DIM; idx += 256) {
        const int i = idx >> 6, j = idx & 63;
        Mre[i * PITCH + j] *= sc;
        Mim[i * PITCH + j] *= sc;
        Rre[i * PITCH + j] = (i == j) ? 1.0f : 0.0f;   // R = I
        Rim[i * PITCH + j] = 0.0f;
    }
    __syncthreads();

    // Horner: R <- I + (M*R)/k,  k = K..1
    for (int k = HORNER_K; k >= 1; --k) {
        cmatmul(Mre, Mim, Rre, Rim, Tre, Tim);
        __syncthreads();
        const float inv = 1.0f / (float)k;
        for (int idx = tid; idx < DIM * DIM; idx += 256) {
            const int i = idx >> 6, j = idx & 63;
            Rre[i * PITCH + j] = ((i == j) ? 1.0f : 0.0f) + Tre[i * PITCH + j] * inv;
            Rim[i * PITCH + j] = Tim[i * PITCH + j] * inv;
        }
        __syncthreads();
    }

    // Squarings with pointer ping-pong (no per-step copy)
    float* Pre = Rre;  float* Pim = Rim;
    float* Qre = Tre;  float* Qim = Tim;
    for (int q = 0; q < nsq; ++q) {
        cmatmul(Pre, Pim, Pre, Pim, Qre, Qim);
        __syncthreads();
        float* t;
        t = Pre; Pre = Qre; Qre = t;
        t = Pim; Pim = Qim; Qim = t;
    }
    if (Pre != Rre) {   // canonicalize result into R buffers (at most one copy)
        for (int idx = tid; idx < DIM * DIM; idx += 256) {
            const int i = idx >> 6, j = idx & 63;
            Rre[i * PITCH + j] = Pre[i * PITCH + j];
            Rim[i * PITCH + j] = Pim[i * PITCH + j];
        }
        __syncthreads();
    }
}

// Kernel 0: build (xmask,zmask) -> pauli-index permutation table.
__global__ void pauli_perm_kernel(unsigned* perm) {
    const int p = blockIdx.x * blockDim.x + threadIdx.x;
    if (p >= NPAULI) return;
    int xm = 0, zm = 0;
    for (int b = 0; b < NQ; ++b) {
        const int d = (p >> (2 * b)) & 3;     // 0=I 1=X 2=Y 3=Z
        xm |= ((d == 1 || d == 2) ? 1 : 0) << b;
        zm |= ((d >= 2) ? 1 : 0) << b;
    }
    perm[xm * DIM + zm] = (unsigned)p;
}

// Kernel 1: Up[r] = expm(-i * sum_p w[r,p-1] P_p), r = blockIdx.x
__global__ void __launch_bounds__(256)
up_kernel(const float* __restrict__ weight, float* __restrict__ upOut,
          const unsigned* __restrict__ perm, int nw) {
    extern __shared__ float lds[];
    const int r = blockIdx.x;
    const int tid = threadIdx.x;
    float* cbuf = lds + OFF_TRE;                   // coefficients staged in T (free now)
    unsigned* permL = (unsigned*)(lds + OFF_RRE);  // perm staged in R (free now)
    for (int p = tid; p < nw; p += 256) cbuf[p] = weight[r * nw + p];
    for (int p = tid; p < NPAULI; p += 256) permL[p] = perm[p];
    __syncthreads();
    build_M(lds, permL, cbuf, /*shift=*/1, nw);
    expm_lds(lds);
    float* Rre = lds + OFF_RRE;  float* Rim = lds + OFF_RIM;
    float* out = upOut + (size_t)r * (2 * DIM * DIM);
    for (int idx = tid; idx < DIM * DIM; idx += 256) {
        const int i = idx >> 6, j = idx & 63;
        out[idx]             = Rre[i * PITCH + j];
        out[DIM * DIM + idx] = Rim[i * PITCH + j];
    }
}

// Kernel 2: per sample b = blockIdx.x : Ud = expm(-iHd), circuit, probs+bias.
__global__ void __launch_bounds__(256)
sample_kernel(const float* __restrict__ x, const float* __restrict__ bias,
              const float* __restrict__ upG, const unsigned* __restrict__ perm,
              float* __restrict__ out, int in_dim, int n_reps) {
    extern __shared__ float lds[];
    const int b = blockIdx.x;
    const int tid = threadIdx.x;
    float* cbuf = lds + OFF_TRE;
    unsigned* permL = (unsigned*)(lds + OFF_RRE);

    // Keep the shared parameter unitaries warm in GL2 for the matvec phase.
    {
        const size_t upBytes = (size_t)n_reps * 2 * DIM * DIM * sizeof(float);
        for (size_t off = (size_t)tid * 64; off < upBytes; off += 256 * 64)
            __builtin_prefetch((const char*)upG + off, 0, 3);
    }

    for (int p = tid; p < in_dim; p += 256)
        cbuf[p] = x[(size_t)b * in_dim + p];
    for (int p = tid; p < NPAULI; p += 256) permL[p] = perm[p];
    __syncthreads();
    build_M(lds, permL, cbuf, /*shift=*/0, in_dim);
    expm_lds(lds);

    float* Ure = lds + OFF_RRE;  float* Uim = lds + OFF_RIM;
    float* scr = lds + OFF_SCR;
    float* psRe = scr + 96;  float* psIm = scr + 160;
    float* p2Re = scr + 224; float* p2Im = scr + 288;
    if (tid < DIM) { psRe[tid] = (tid == 0) ? 1.0f : 0.0f; psIm[tid] = 0.0f; }
    __syncthreads();

    for (int r = 0; r < n_reps; ++r) {
        if (tid < DIM) {                    // psi2 = Ud * psi (LDS matvec)
            float ar = 0.f, ai = 0.f;
            for (int j = 0; j < DIM; ++j) {
                const float ur = Ure[tid * PITCH + j], ui = Uim[tid * PITCH + j];
                const float pr = psRe[j], pi = psIm[j];
                ar += ur * pr - ui * pi;
                ai += ur * pi + ui * pr;
            }
            p2Re[tid] = ar; p2Im[tid] = ai;
        }
        __syncthreads();
        if (tid < DIM) {                    // psi = Up[r] * psi2 (global matvec)
            const float* gr = upG + (size_t)r * (2 * DIM * DIM);
            const float* gi = gr + DIM * DIM;
            float ar = 0.f, ai = 0.f;
            for (int j = 0; j < DIM; ++j) {
                const float ur = gr[tid * DIM + j], ui = gi[tid * DIM + j];
                const float pr = p2Re[j], pi = p2Im[j];
                ar += ur * pr - ui * pi;
                ai += ur * pi + ui * pr;
            }
            psRe[tid] = ar; psIm[tid] = ai;
        }
        __syncthreads();
    }
    if (tid < DIM)
        out[(size_t)b * DIM + tid] = psRe[tid] * psRe[tid] + psIm[tid] * psIm[tid] + bias[tid];
}

extern "C" void kernel_launch(void* const* d_in, const int* in_sizes, int n_in,
                              void* d_out, int out_size, void* d_ws, size_t ws_size,
                              hipStream_t stream) {
    const float* x      = (const float*)d_in[0];
    const float* weight = (const float*)d_in[1];
    const float* bias   = (const float*)d_in[2];
    float* out          = (float*)d_out;

    const int in_dim = 4000;
    const int B      = in_sizes[0] / in_dim;           // 512
    const int nw     = NPAULI - 1;                     // 4095
    const int n_reps = in_sizes[1] / nw;               // 4

    unsigned* perm = (unsigned*)d_ws;
    float*    upG  = (float*)((char*)d_ws + NPAULI * sizeof(unsigned));

    const size_t ldsBytes = LDS_FLOATS * sizeof(float);

    pauli_perm_kernel<<<NPAULI / 256, 256, 0, stream>>>(perm);
    up_kernel<<<n_reps, 256, ldsBytes, stream>>>(weight, upG, perm, nw);
    sample_kernel<<<B, 256, ldsBytes, stream>>>(x, bias, upG, perm, out, in_dim, n_reps);
}